// RecurrentGCN_27745488732311
// MI455X (gfx1250) — compile-verified
//
#include <hip/hip_runtime.h>
#include <math.h>

typedef float v2f __attribute__((ext_vector_type(2)));
typedef float v8f __attribute__((ext_vector_type(8)));

#define NN 50000
#define NE 800000
#define TSTEPS 12
#define HID 64
#define NOUT 12

__device__ __forceinline__ float sigmoidf_(float x) {
    return 1.0f / (1.0f + __expf(-x));
}

// ---------------------------------------------------------------------------
// degree + norm precompute
// ---------------------------------------------------------------------------
__global__ __launch_bounds__(256) void deg_kernel(const int* __restrict__ row,
                                                  const int* __restrict__ col,
                                                  const float* __restrict__ w,
                                                  float* __restrict__ deg_out,
                                                  float* __restrict__ deg_in) {
    int e = blockIdx.x * 256 + threadIdx.x;
    if (e >= NE) return;
    unsafeAtomicAdd(deg_out + row[e], w[e]);
    unsafeAtomicAdd(deg_in + col[e], w[e]);
}

__global__ __launch_bounds__(256) void norm_kernel(const int* __restrict__ row,
                                                   const int* __restrict__ col,
                                                   const float* __restrict__ w,
                                                   const float* __restrict__ deg_out,
                                                   const float* __restrict__ deg_in,
                                                   float* __restrict__ no,
                                                   float* __restrict__ ni) {
    int e = blockIdx.x * 256 + threadIdx.x;
    if (e >= NE) return;
    float we = w[e];
    no[e] = we / deg_out[row[e]];
    ni[e] = we / deg_in[col[e]];
}

// ---------------------------------------------------------------------------
// weight packing: src shape (2, 2, c, 64) flat ((d*2+k)*c + i)*64 + j
// sel 0: w[0,0]+w[1,0]  (self term, K folded)
// sel 1: w[0,1]         (out-diffusion)
// sel 2: w[1,1]         (in-diffusion)
// dst is (c_pad x ncol) row-major; for ncol==128, cols 0..63 from wA, 64..127 from wB
// ---------------------------------------------------------------------------
__global__ __launch_bounds__(256) void pack_kernel(float* __restrict__ dst,
                                                   const float* __restrict__ wA,
                                                   const float* __restrict__ wB,
                                                   int c, int c_pad, int sel, int ncol) {
    int idx = blockIdx.x * 256 + threadIdx.x;
    if (idx >= c_pad * ncol) return;
    int k = idx / ncol;
    int j = idx - k * ncol;
    const float* w = (j < 64) ? wA : wB;
    int jj = j & 63;
    float v = 0.0f;
    if (k < c) {
        if (sel == 0)      v = w[(0 * c + k) * 64 + jj] + w[(2 * c + k) * 64 + jj];
        else if (sel == 1) v = w[(1 * c + k) * 64 + jj];
        else               v = w[(3 * c + k) * 64 + jj];
    }
    dst[idx] = v;
}

// ---------------------------------------------------------------------------
// XH builders
// ---------------------------------------------------------------------------
__global__ __launch_bounds__(256) void build_xh1_kernel(float* __restrict__ XH,
                                                        const float* __restrict__ x,
                                                        const float* __restrict__ H1,
                                                        int t) {
    int idx = blockIdx.x * 256 + threadIdx.x;
    if (idx >= NN * 68) return;
    int n = idx / 68;
    int c = idx - n * 68;
    float v;
    if (c < 2)       v = x[n * 24 + c * 12 + t];     // x[1,N,2,12]
    else if (c < 66) v = H1[n * 64 + (c - 2)];
    else             v = 0.0f;                        // pad cols
    XH[idx] = v;
}

__global__ __launch_bounds__(256) void build_xh2_kernel(float* __restrict__ XH,
                                                        const float* __restrict__ H1,
                                                        const float* __restrict__ H2) {
    int idx = blockIdx.x * 256 + threadIdx.x;
    if (idx >= NN * 128) return;
    int n = idx >> 7;
    int c = idx & 127;
    XH[idx] = (c < 64) ? H1[n * 64 + c] : H2[n * 64 + (c - 64)];
}

// ---------------------------------------------------------------------------
// edge scatter: Tx_o += no[e]*XH[row], Tx_i += ni[e]*XH[row], indexed by col
// one thread per (edge, 4-feature chunk)
// ---------------------------------------------------------------------------
template <int NCHUNK, int CPAD>
__global__ __launch_bounds__(256) void scatter_kernel(const float* __restrict__ XH,
                                                      float* __restrict__ Txo,
                                                      float* __restrict__ Txi,
                                                      const int* __restrict__ row,
                                                      const int* __restrict__ col,
                                                      const float* __restrict__ no,
                                                      const float* __restrict__ ni) {
    int idx = blockIdx.x * 256 + threadIdx.x;
    if (idx >= NE * NCHUNK) return;
    int e  = idx / NCHUNK;
    int f4 = (idx - e * NCHUNK) * 4;
    int r  = row[e];
    int cn = col[e];
    float a = no[e];
    float b = ni[e];
    const float4 xv = *(const float4*)(XH + (size_t)r * CPAD + f4);
    float* po = Txo + (size_t)cn * CPAD + f4;
    float* pi = Txi + (size_t)cn * CPAD + f4;
    unsafeAtomicAdd(po + 0, a * xv.x);
    unsafeAtomicAdd(po + 1, a * xv.y);
    unsafeAtomicAdd(po + 2, a * xv.z);
    unsafeAtomicAdd(po + 3, a * xv.w);
    unsafeAtomicAdd(pi + 0, b * xv.x);
    unsafeAtomicAdd(pi + 1, b * xv.y);
    unsafeAtomicAdd(pi + 2, b * xv.z);
    unsafeAtomicAdd(pi + 3, b * xv.w);
}

// ---------------------------------------------------------------------------
// WMMA fp32 GEMM core: one wave computes a 16-row tile x (NG*16) cols of
//   [XH | Txo | Txi] @ [Ws ; Wo ; Wi]
// A-frag (16x4 f32): lanes 0-15 rows M=lane K={k0,k0+1}; lanes 16-31 K={k0+2,k0+3}
// B-frag mirrored with N=lane&15. C/D: M = v + 8*(lane>=16), N = lane&15.
// ---------------------------------------------------------------------------
template <int NG>
__device__ __forceinline__ void gemm_core(v8f acc[NG],
                                          const float* __restrict__ XH,
                                          const float* __restrict__ Txo,
                                          const float* __restrict__ Txi,
                                          const float* __restrict__ Ws,
                                          const float* __restrict__ Wo,
                                          const float* __restrict__ Wi,
                                          int c_pad, int tile) {
    const int lane = threadIdx.x & 31;
    const int half = lane >> 4;
    const int lrow = lane & 15;
    const size_t rowoff = (size_t)(tile * 16 + lrow) * c_pad + 2 * half;
    const float* Aseg[3] = {XH + rowoff, Txo + rowoff, Txi + rowoff};
    const float* Wseg[3] = {Ws, Wo, Wi};
    const int ncol = NG * 16;
    for (int s = 0; s < 3; ++s) {
        const float* A = Aseg[s];
        const float* W = Wseg[s] + (size_t)(2 * half) * ncol + lrow;
        for (int k0 = 0; k0 < c_pad; k0 += 4) {
            v2f a = *(const v2f*)(A + k0);
            const float* Wk = W + (size_t)k0 * ncol;
#pragma unroll
            for (int g = 0; g < NG; ++g) {
                v2f b;
                b.x = Wk[g * 16];
                b.y = Wk[ncol + g * 16];
                acc[g] = __builtin_amdgcn_wmma_f32_16x16x4_f32(
                    false, a, false, b, (short)0, acc[g], false, false);
            }
        }
    }
}

// Z/R gates fused: cols 0..63 -> Z = sigmoid(.), stored; cols 64..127 -> R,
// immediately folded as XH[n, F+j] = R * H[n,j]  (builds cat([X, R*H]) in place).
// NOTE: XH and XHdst alias (same buffer) -> no __restrict__ on them.
__global__ __launch_bounds__(128) void gemm_zr_kernel(const float* XH,
                                                      const float* __restrict__ Txo,
                                                      const float* __restrict__ Txi,
                                                      const float* __restrict__ Ws,
                                                      const float* __restrict__ Wo,
                                                      const float* __restrict__ Wi,
                                                      const float* __restrict__ bz,
                                                      const float* __restrict__ br,
                                                      float* __restrict__ Zbuf,
                                                      const float* __restrict__ H,
                                                      float* XHdst,
                                                      int F, int c_pad, int ntiles) {
    int tile = blockIdx.x * 4 + (threadIdx.x >> 5);
    if (tile >= ntiles) return;
    v8f acc[8];
    const v8f vz = {0, 0, 0, 0, 0, 0, 0, 0};
#pragma unroll
    for (int g = 0; g < 8; ++g) acc[g] = vz;
    gemm_core<8>(acc, XH, Txo, Txi, Ws, Wo, Wi, c_pad, tile);

    const int lane = threadIdx.x & 31;
    const int half = lane >> 4;
    const int lrow = lane & 15;
#pragma unroll
    for (int g = 0; g < 4; ++g) {
        int j = g * 16 + lrow;
        float bj = bz[j];
#pragma unroll
        for (int v = 0; v < 8; ++v) {
            int n = tile * 16 + v + 8 * half;
            Zbuf[(size_t)n * 64 + j] = sigmoidf_(acc[g][v] + bj);
        }
    }
#pragma unroll
    for (int g = 4; g < 8; ++g) {
        int j = (g - 4) * 16 + lrow;
        float bj = br[j];
#pragma unroll
        for (int v = 0; v < 8; ++v) {
            int n = tile * 16 + v + 8 * half;
            float r = sigmoidf_(acc[g][v] + bj);
            XHdst[(size_t)n * c_pad + F + j] = r * H[(size_t)n * 64 + j];
        }
    }
}

// candidate gate fused with GRU blend + relu, H updated in place
__global__ __launch_bounds__(128) void gemm_h_kernel(const float* __restrict__ XH,
                                                     const float* __restrict__ Txo,
                                                     const float* __restrict__ Txi,
                                                     const float* __restrict__ Ws,
                                                     const float* __restrict__ Wo,
                                                     const float* __restrict__ Wi,
                                                     const float* __restrict__ bh,
                                                     const float* __restrict__ Zbuf,
                                                     float* __restrict__ H,
                                                     int c_pad, int ntiles) {
    int tile = blockIdx.x * 4 + (threadIdx.x >> 5);
    if (tile >= ntiles) return;
    v8f acc[4];
    const v8f vz = {0, 0, 0, 0, 0, 0, 0, 0};
#pragma unroll
    for (int g = 0; g < 4; ++g) acc[g] = vz;
    gemm_core<4>(acc, XH, Txo, Txi, Ws, Wo, Wi, c_pad, tile);

    const int lane = threadIdx.x & 31;
    const int half = lane >> 4;
    const int lrow = lane & 15;
#pragma unroll
    for (int g = 0; g < 4; ++g) {
        int j = g * 16 + lrow;
        float bj = bh[j];
#pragma unroll
        for (int v = 0; v < 8; ++v) {
            int n = tile * 16 + v + 8 * half;
            float ht = tanhf(acc[g][v] + bj);
            float z = Zbuf[(size_t)n * 64 + j];
            float h = H[(size_t)n * 64 + j];
            float hn = z * h + (1.0f - z) * ht;
            H[(size_t)n * 64 + j] = fmaxf(hn, 0.0f);
        }
    }
}

// ---------------------------------------------------------------------------
// final projection out = H2 @ lin_w + lin_b   (N x 64 @ 64 x 12)
// ---------------------------------------------------------------------------
__global__ __launch_bounds__(256) void final_kernel(const float* __restrict__ H2,
                                                    const float* __restrict__ lw,
                                                    const float* __restrict__ lb,
                                                    float* __restrict__ out) {
    int idx = blockIdx.x * 256 + threadIdx.x;
    if (idx >= NN * NOUT) return;
    int n = idx / NOUT;
    int o = idx - n * NOUT;
    float s = lb[o];
    const float* h = H2 + (size_t)n * 64;
#pragma unroll
    for (int j = 0; j < 64; ++j) s += h[j] * lw[j * NOUT + o];
    out[idx] = s;
}

// ---------------------------------------------------------------------------
extern "C" void kernel_launch(void* const* d_in, const int* in_sizes, int n_in,
                              void* d_out, int out_size, void* d_ws, size_t ws_size,
                              hipStream_t stream) {
    const float* x   = (const float*)d_in[0];
    const int*   ei  = (const int*)d_in[1];
    const int*   row = ei;
    const int*   col = ei + NE;
    const float* ew  = (const float*)d_in[2];
    const float* w1z = (const float*)d_in[3];
    const float* b1z = (const float*)d_in[4];
    const float* w1r = (const float*)d_in[5];
    const float* b1r = (const float*)d_in[6];
    const float* w1h = (const float*)d_in[7];
    const float* b1h = (const float*)d_in[8];
    const float* w2z = (const float*)d_in[9];
    const float* b2z = (const float*)d_in[10];
    const float* w2r = (const float*)d_in[11];
    const float* b2r = (const float*)d_in[12];
    const float* w2h = (const float*)d_in[13];
    const float* b2h = (const float*)d_in[14];
    const float* lw  = (const float*)d_in[15];
    const float* lb  = (const float*)d_in[16];
    float* out = (float*)d_out;

    // workspace layout (~123 MB total)
    float* ws = (float*)d_ws;
    size_t off = 0;
    float* deg_out = ws + off; off += NN;
    float* deg_in  = ws + off; off += NN;
    float* nrm_o   = ws + off; off += NE;
    float* nrm_i   = ws + off; off += NE;
    float* H1      = ws + off; off += (size_t)NN * 64;
    float* H2      = ws + off; off += (size_t)NN * 64;
    float* Zb      = ws + off; off += (size_t)NN * 64;
    float* XH      = ws + off; off += (size_t)NN * 128;
    float* Txo     = ws + off; off += (size_t)NN * 128;
    float* Txi     = ws + off; off += (size_t)NN * 128;
    float* W1      = ws + off; off += (size_t)68 * 576;
    float* W2      = ws + off; off += (size_t)128 * 576;

    // per-layer packed weight sub-buffers: [ZRself | ZRout | ZRin | Hself | Hout | Hin]
    float* Wzr_s1 = W1;                 float* Wzr_o1 = W1 + 68 * 128;
    float* Wzr_i1 = W1 + 2 * 68 * 128;  float* Wh_s1  = W1 + 3 * 68 * 128;
    float* Wh_o1  = Wh_s1 + 68 * 64;    float* Wh_i1  = Wh_s1 + 2 * 68 * 64;
    float* Wzr_s2 = W2;                 float* Wzr_o2 = W2 + 128 * 128;
    float* Wzr_i2 = W2 + 2 * 128 * 128; float* Wh_s2  = W2 + 3 * 128 * 128;
    float* Wh_o2  = Wh_s2 + 128 * 64;   float* Wh_i2  = Wh_s2 + 2 * 128 * 64;

    const int ntiles = NN / 16;           // 3125
    const int gemmBlocks = (ntiles + 3) / 4;

    // init
    hipMemsetAsync(deg_out, 0, 2 * NN * sizeof(float), stream);   // deg_out+deg_in
    hipMemsetAsync(H1, 0, 2 * (size_t)NN * 64 * sizeof(float), stream); // H1+H2

    deg_kernel<<<(NE + 255) / 256, 256, 0, stream>>>(row, col, ew, deg_out, deg_in);
    norm_kernel<<<(NE + 255) / 256, 256, 0, stream>>>(row, col, ew, deg_out, deg_in, nrm_o, nrm_i);

    // pack weights (padded c=66 -> 68 for layer 1)
    {
        int g1 = (68 * 128 + 255) / 256, g1h = (68 * 64 + 255) / 256;
        pack_kernel<<<g1, 256, 0, stream>>>(Wzr_s1, w1z, w1r, 66, 68, 0, 128);
        pack_kernel<<<g1, 256, 0, stream>>>(Wzr_o1, w1z, w1r, 66, 68, 1, 128);
        pack_kernel<<<g1, 256, 0, stream>>>(Wzr_i1, w1z, w1r, 66, 68, 2, 128);
        pack_kernel<<<g1h, 256, 0, stream>>>(Wh_s1, w1h, w1h, 66, 68, 0, 64);
        pack_kernel<<<g1h, 256, 0, stream>>>(Wh_o1, w1h, w1h, 66, 68, 1, 64);
        pack_kernel<<<g1h, 256, 0, stream>>>(Wh_i1, w1h, w1h, 66, 68, 2, 64);
        int g2 = (128 * 128 + 255) / 256, g2h = (128 * 64 + 255) / 256;
        pack_kernel<<<g2, 256, 0, stream>>>(Wzr_s2, w2z, w2r, 128, 128, 0, 128);
        pack_kernel<<<g2, 256, 0, stream>>>(Wzr_o2, w2z, w2r, 128, 128, 1, 128);
        pack_kernel<<<g2, 256, 0, stream>>>(Wzr_i2, w2z, w2r, 128, 128, 2, 128);
        pack_kernel<<<g2h, 256, 0, stream>>>(Wh_s2, w2h, w2h, 128, 128, 0, 64);
        pack_kernel<<<g2h, 256, 0, stream>>>(Wh_o2, w2h, w2h, 128, 128, 1, 64);
        pack_kernel<<<g2h, 256, 0, stream>>>(Wh_i2, w2h, w2h, 128, 128, 2, 64);
    }

    const size_t tx1Bytes = (size_t)NN * 68 * sizeof(float);
    const size_t tx2Bytes = (size_t)NN * 128 * sizeof(float);
    const int sc1Grid = (NE * 17 + 255) / 256;
    const int sc2Grid = (NE * 32 + 255) / 256;

    for (int t = 0; t < TSTEPS; ++t) {
        // ---- layer 1 (c=66 padded to 68, F=2, state H1) ----
        build_xh1_kernel<<<(NN * 68 + 255) / 256, 256, 0, stream>>>(XH, x, H1, t);
        hipMemsetAsync(Txo, 0, tx1Bytes, stream);
        hipMemsetAsync(Txi, 0, tx1Bytes, stream);
        scatter_kernel<17, 68><<<sc1Grid, 256, 0, stream>>>(XH, Txo, Txi, row, col, nrm_o, nrm_i);
        gemm_zr_kernel<<<gemmBlocks, 128, 0, stream>>>(XH, Txo, Txi, Wzr_s1, Wzr_o1, Wzr_i1,
                                                       b1z, b1r, Zb, H1, XH, 2, 68, ntiles);
        hipMemsetAsync(Txo, 0, tx1Bytes, stream);
        hipMemsetAsync(Txi, 0, tx1Bytes, stream);
        scatter_kernel<17, 68><<<sc1Grid, 256, 0, stream>>>(XH, Txo, Txi, row, col, nrm_o, nrm_i);
        gemm_h_kernel<<<gemmBlocks, 128, 0, stream>>>(XH, Txo, Txi, Wh_s1, Wh_o1, Wh_i1,
                                                      b1h, Zb, H1, 68, ntiles);

        // ---- layer 2 (c=128, F=64, state H2, input H1) ----
        build_xh2_kernel<<<(NN * 128 + 255) / 256, 256, 0, stream>>>(XH, H1, H2);
        hipMemsetAsync(Txo, 0, tx2Bytes, stream);
        hipMemsetAsync(Txi, 0, tx2Bytes, stream);
        scatter_kernel<32, 128><<<sc2Grid, 256, 0, stream>>>(XH, Txo, Txi, row, col, nrm_o, nrm_i);
        gemm_zr_kernel<<<gemmBlocks, 128, 0, stream>>>(XH, Txo, Txi, Wzr_s2, Wzr_o2, Wzr_i2,
                                                       b2z, b2r, Zb, H2, XH, 64, 128, ntiles);
        hipMemsetAsync(Txo, 0, tx2Bytes, stream);
        hipMemsetAsync(Txi, 0, tx2Bytes, stream);
        scatter_kernel<32, 128><<<sc2Grid, 256, 0, stream>>>(XH, Txo, Txi, row, col, nrm_o, nrm_i);
        gemm_h_kernel<<<gemmBlocks, 128, 0, stream>>>(XH, Txo, Txi, Wh_s2, Wh_o2, Wh_i2,
                                                      b2h, Zb, H2, 128, ntiles);
    }

    final_kernel<<<(NN * NOUT + 255) / 256, 256, 0, stream>>>(H2, lw, lb, out);
}